// EquiformerV2ForceHeadDenoising_43928925503854
// MI455X (gfx1250) — compile-verified
//
#include <hip/hip_runtime.h>

// ---------------------------------------------------------------------------
// EquiformerV2 force-head + denoising head, CDNA5 (gfx1250), fp32 WMMA path.
// N=4000 nodes, E=60000 edges, M=16, C=H=64, HEADS=8, A=VC=8, CE=32.
// ---------------------------------------------------------------------------

typedef __attribute__((ext_vector_type(2))) float v2f;
typedef __attribute__((ext_vector_type(8))) float v8f;

__device__ __forceinline__ v8f wmma4(v2f a, v2f b, v8f c) {
  // D = A(16x4,f32) * B(4x16,f32) + C(16x16,f32)
  return __builtin_amdgcn_wmma_f32_16x16x4_f32(false, a, false, b, (short)0, c,
                                               false, false);
}

// monotone float<->uint encoding for atomicMax on floats
__device__ __forceinline__ unsigned fenc(float f) {
  unsigned u = __float_as_uint(f);
  return (u & 0x80000000u) ? ~u : (u | 0x80000000u);
}
__device__ __forceinline__ float fdec(unsigned k) {
  unsigned u = (k & 0x80000000u) ? (k & 0x7fffffffu) : ~k;
  return __uint_as_float(u);
}
__device__ __forceinline__ float silu_f(float v) { return v / (1.f + __expf(-v)); }
__device__ __forceinline__ float lrelu_f(float v) { return v > 0.f ? v : 0.2f * v; }

struct BlockW {  // per-head weight set (fs_* or dn_*)
  const float *es, *et, *wd, *We, *Ws, *Wt, *Wa, *va, *Wv, *Wo;
};

// ---------------------------------------------------------------------------
// Pass 0: zero workspace accumulators (graph replays must be self-initializing)
// ---------------------------------------------------------------------------
__global__ void init_ws(unsigned* nm_fs, unsigned* nm_dn, float* ns_fs,
                        float* ns_dn, float* ac_fs, float* ac_dn, int n8, int n3) {
  int i = blockIdx.x * 256 + threadIdx.x;
  if (i < n8) { nm_fs[i] = 0u; nm_dn[i] = 0u; ns_fs[i] = 0.f; ns_dn[i] = 0.f; }
  if (i < n3) { ac_fs[i] = 0.f; ac_dn[i] = 0.f; }
}

// ---------------------------------------------------------------------------
// Pass 1: per-edge attention logits (needs only m=0 row). Wave per edge,
// all cross-lane traffic via wave32 shuffles (no LDS, no barriers).
// ---------------------------------------------------------------------------
__global__ __launch_bounds__(256) void pass_logits(
    const float* __restrict__ x, const int* __restrict__ zn,
    const float* __restrict__ dist, const int* __restrict__ eidx,
    const float* __restrict__ wig, BlockW fs, BlockW dn,
    float* __restrict__ lg_fs, float* __restrict__ lg_dn,
    unsigned* __restrict__ nm_fs, unsigned* __restrict__ nm_dn, int E_) {
  int wave = threadIdx.x >> 5, lane = threadIdx.x & 31;
  int e = blockIdx.x * 8 + wave;
  if (e >= E_) return;  // wave-uniform exit
  int src = eidx[e], dst = eidx[E_ + e];
  const float* Dg = wig + (size_t)e * 256;
  const float* Xs = x + (size_t)src * 1024;
  const float* Xt = x + (size_t)dst * 1024;
  int c1 = lane, c2 = lane + 32;

  // row-0 rotation: xs0[c] = sum_n D[0][n] * x[src][n][c]
  float xs1 = 0.f, xs2 = 0.f, xt1 = 0.f, xt2 = 0.f;
  for (int n = 0; n < 16; ++n) {
    float d = Dg[n];
    xs1 += d * Xs[n * 64 + c1]; xs2 += d * Xs[n * 64 + c2];
    xt1 += d * Xt[n * 64 + c1]; xt2 += d * Xt[n * 64 + c2];
  }
  // edge scalar embedding, lane j holds e[j] (CE==32==wave)
  float de = dist[e];
  int zs = zn[src], zt = zn[dst];
  float efs = silu_f(de * fs.wd[lane] + fs.es[zs * 32 + lane] + fs.et[zt * 32 + lane]);
  float edn = silu_f(de * dn.wd[lane] + dn.es[zs * 32 + lane] + dn.et[zt * 32 + lane]);

  // f0 = xs0@Ws + xt0@Wt + e@We   (both heads, 2 columns per lane)
  float ffs1 = 0.f, ffs2 = 0.f, fdn1 = 0.f, fdn2 = 0.f;
  for (int cc = 0; cc < 64; ++cc) {
    float xsv = __shfl(cc < 32 ? xs1 : xs2, cc & 31, 32);
    float xtv = __shfl(cc < 32 ? xt1 : xt2, cc & 31, 32);
    ffs1 += xsv * fs.Ws[cc * 64 + c1] + xtv * fs.Wt[cc * 64 + c1];
    ffs2 += xsv * fs.Ws[cc * 64 + c2] + xtv * fs.Wt[cc * 64 + c2];
    fdn1 += xsv * dn.Ws[cc * 64 + c1] + xtv * dn.Wt[cc * 64 + c1];
    fdn2 += xsv * dn.Ws[cc * 64 + c2] + xtv * dn.Wt[cc * 64 + c2];
  }
  for (int j = 0; j < 32; ++j) {
    float ef = __shfl(efs, j, 32), ed = __shfl(edn, j, 32);
    ffs1 += ef * fs.We[j * 64 + c1]; ffs2 += ef * fs.We[j * 64 + c2];
    fdn1 += ed * dn.We[j * 64 + c1]; fdn2 += ed * dn.We[j * 64 + c2];
  }
  // a = leaky_relu(f0 @ Wa)
  float afs1 = 0.f, afs2 = 0.f, adn1 = 0.f, adn2 = 0.f;
  for (int cc = 0; cc < 64; ++cc) {
    float vfs = __shfl(cc < 32 ? ffs1 : ffs2, cc & 31, 32);
    float vdn = __shfl(cc < 32 ? fdn1 : fdn2, cc & 31, 32);
    afs1 += vfs * fs.Wa[cc * 64 + c1]; afs2 += vfs * fs.Wa[cc * 64 + c2];
    adn1 += vdn * dn.Wa[cc * 64 + c1]; adn2 += vdn * dn.Wa[cc * 64 + c2];
  }
  float pfs1 = lrelu_f(afs1) * fs.va[c1], pfs2 = lrelu_f(afs2) * fs.va[c2];
  float pdn1 = lrelu_f(adn1) * dn.va[c1], pdn2 = lrelu_f(adn2) * dn.va[c2];
  // sum over A=8 within head (groups of 8 lanes)
  for (int m = 1; m <= 4; m <<= 1) {
    pfs1 += __shfl_xor(pfs1, m, 32); pfs2 += __shfl_xor(pfs2, m, 32);
    pdn1 += __shfl_xor(pdn1, m, 32); pdn2 += __shfl_xor(pdn2, m, 32);
  }
  if ((lane & 7) == 0) {
    int h1 = lane >> 3, h2 = 4 + (lane >> 3);
    lg_fs[(size_t)e * 8 + h1] = pfs1; lg_fs[(size_t)e * 8 + h2] = pfs2;
    lg_dn[(size_t)e * 8 + h1] = pdn1; lg_dn[(size_t)e * 8 + h2] = pdn2;
    atomicMax(&nm_fs[dst * 8 + h1], fenc(pfs1));
    atomicMax(&nm_fs[dst * 8 + h2], fenc(pfs2));
    atomicMax(&nm_dn[dst * 8 + h1], fenc(pdn1));
    atomicMax(&nm_dn[dst * 8 + h2], fenc(pdn2));
  }
}

// ---------------------------------------------------------------------------
// Pass 2: ex = exp(logit - segmax); accumulate per-node sums.
// ---------------------------------------------------------------------------
__global__ __launch_bounds__(256) void pass_soft(
    const int* __restrict__ eidx, float* __restrict__ lg_fs,
    float* __restrict__ lg_dn, const unsigned* __restrict__ nm_fs,
    const unsigned* __restrict__ nm_dn, float* __restrict__ ns_fs,
    float* __restrict__ ns_dn, int E_) {
  int i = blockIdx.x * 256 + threadIdx.x;
  if (i >= E_ * 8) return;
  int e = i >> 3, h = i & 7;
  int dst = eidx[E_ + e];
  float ex = __expf(lg_fs[i] - fdec(nm_fs[dst * 8 + h]));
  lg_fs[i] = ex;
  atomicAdd(&ns_fs[dst * 8 + h], ex);
  ex = __expf(lg_dn[i] - fdec(nm_dn[dst * 8 + h]));
  lg_dn[i] = ex;
  atomicAdd(&ns_dn[dst * 8 + h], ex);
}

// ---------------------------------------------------------------------------
// Pass 3: heavy per-edge GEMMs via V_WMMA_F32_16X16X4_F32.
// One workgroup (4 waves) per edge; wave t owns output columns 16t..16t+15.
// xs/xt computed once, reused by both heads.
// ---------------------------------------------------------------------------
__global__ __launch_bounds__(128) void pass_main(
    const float* __restrict__ x, const int* __restrict__ zn,
    const float* __restrict__ dist, const int* __restrict__ eidx,
    const float* __restrict__ wig, BlockW fs, BlockW dn,
    const float* __restrict__ ex_fs, const float* __restrict__ ex_dn,
    const float* __restrict__ ns_fs, const float* __restrict__ ns_dn,
    float* __restrict__ ac_fs, float* __restrict__ ac_dn, int E_) {
  __shared__ float xs_lds[16 * 64];
  __shared__ float xt_lds[16 * 64];
  __shared__ float f_lds[16 * 64];
  __shared__ float e_lds[32];

  int e = blockIdx.x;
  int tid = threadIdx.x;
  int wave = tid >> 5, lane = tid & 31, half = lane >> 4, lm = lane & 15;
  int colg = wave * 16 + lm;  // this lane's output column (0..63)
  int src = eidx[e], dst = eidx[E_ + e];
  const float* Dg = wig + (size_t)e * 256;   // D[m][n] row-major 16x16
  const float* Xs = x + (size_t)src * 1024;  // [16][64]
  const float* Xt = x + (size_t)dst * 1024;

  // --- GEMM1: xs = D @ x[src], xt = D @ x[dst]; wave computes its 16-col tile
  v8f accS = {0.f, 0.f, 0.f, 0.f, 0.f, 0.f, 0.f, 0.f};
  v8f accT = {0.f, 0.f, 0.f, 0.f, 0.f, 0.f, 0.f, 0.f};
#pragma unroll
  for (int kc = 0; kc < 4; ++kc) {
    int k0 = kc * 4 + half * 2;  // K chunk of 4, split across lane halves
    v2f a = {Dg[lm * 16 + k0], Dg[lm * 16 + k0 + 1]};
    v2f bs = {Xs[k0 * 64 + colg], Xs[(k0 + 1) * 64 + colg]};
    v2f bt = {Xt[k0 * 64 + colg], Xt[(k0 + 1) * 64 + colg]};
    accS = wmma4(a, bs, accS);
    accT = wmma4(a, bt, accT);
  }
#pragma unroll
  for (int r = 0; r < 8; ++r) {  // C/D layout: row = r + 8*half, col = colg
    xs_lds[(r + 8 * half) * 64 + colg] = accS[r];
    xt_lds[(r + 8 * half) * 64 + colg] = accT[r];
  }
  __syncthreads();

  for (int b = 0; b < 2; ++b) {
    const BlockW W = b ? dn : fs;
    const float* exb = b ? ex_dn : ex_fs;
    const float* nsb = b ? ns_dn : ns_fs;
    float* acb = b ? ac_dn : ac_fs;

    __syncthreads();  // previous iteration's f_lds/e_lds readers are done
    if (tid < 32) {
      float v = dist[e] * W.wd[tid] + W.es[zn[src] * 32 + tid] +
                W.et[zn[dst] * 32 + tid];
      e_lds[tid] = silu_f(v);
    }
    __syncthreads();

    // --- GEMM2: f = xs@Ws + xt@Wt  (K=64, 16 chunks each)
    v8f accF = {0.f, 0.f, 0.f, 0.f, 0.f, 0.f, 0.f, 0.f};
#pragma unroll
    for (int kc = 0; kc < 16; ++kc) {
      int k0 = kc * 4 + half * 2;
      v2f aS = {xs_lds[lm * 64 + k0], xs_lds[lm * 64 + k0 + 1]};
      v2f bS = {W.Ws[k0 * 64 + colg], W.Ws[(k0 + 1) * 64 + colg]};
      accF = wmma4(aS, bS, accF);
      v2f aT = {xt_lds[lm * 64 + k0], xt_lds[lm * 64 + k0 + 1]};
      v2f bT = {W.Wt[k0 * 64 + colg], W.Wt[(k0 + 1) * 64 + colg]};
      accF = wmma4(aT, bT, accF);
    }
    // m=0 row inject: f[0,:] += e @ We  (row 0 lives in reg 0 of half 0)
    if (half == 0) {
      float inj = 0.f;
      for (int j = 0; j < 32; ++j) inj += e_lds[j] * W.We[j * 64 + colg];
      accF[0] += inj;
    }
#pragma unroll
    for (int r = 0; r < 8; ++r) f_lds[(r + 8 * half) * 64 + colg] = accF[r];
    __syncthreads();

    // alpha for this lane's column (head = col/8)
    int h = colg >> 3;
    float alpha = exb[(size_t)e * 8 + h] / (nsb[dst * 8 + h] + 1e-9f);

    // --- GEMM3: v = f @ Wv  (K=64)
    v8f accV = {0.f, 0.f, 0.f, 0.f, 0.f, 0.f, 0.f, 0.f};
#pragma unroll
    for (int kc = 0; kc < 16; ++kc) {
      int k0 = kc * 4 + half * 2;
      v2f aF = {f_lds[lm * 64 + k0], f_lds[lm * 64 + k0 + 1]};
      v2f bV = {W.Wv[k0 * 64 + colg], W.Wv[(k0 + 1) * 64 + colg]};
      accV = wmma4(aF, bV, accV);
    }

    // fold Wo then D columns 1..3:  out[m] += sum_k D[k][m] * (v[k,:]@Wo)
    float wo = W.Wo[colg];
    float p[8];
#pragma unroll
    for (int r = 0; r < 8; ++r) p[r] = accV[r] * alpha * wo;
#pragma unroll
    for (int m = 1; m <= 8; m <<= 1) {
#pragma unroll
      for (int r = 0; r < 8; ++r) p[r] += __shfl_xor(p[r], m, 32);
    }
    // p[r] now = partial w[8*half + r] over this wave's 16 columns
    float s0 = 0.f, s1 = 0.f, s2 = 0.f;
#pragma unroll
    for (int r = 0; r < 8; ++r) {
      int k = 8 * half + r;
      s0 += p[r] * Dg[k * 16 + 1];
      s1 += p[r] * Dg[k * 16 + 2];
      s2 += p[r] * Dg[k * 16 + 3];
    }
    s0 += __shfl_xor(s0, 16, 32);
    s1 += __shfl_xor(s1, 16, 32);
    s2 += __shfl_xor(s2, 16, 32);
    if (lane == 0) {
      atomicAdd(&acb[dst * 3 + 0], s0);
      atomicAdd(&acb[dst * 3 + 1], s1);
      atomicAdd(&acb[dst * 3 + 2], s2);
    }
  }
}

// ---------------------------------------------------------------------------
// Pass 4: mask mix -> d_out [N,3]
// ---------------------------------------------------------------------------
__global__ void pass_out(const unsigned char* __restrict__ mask,
                         const float* __restrict__ ac_fs,
                         const float* __restrict__ ac_dn,
                         float* __restrict__ out, int N_) {
  int i = blockIdx.x * 256 + threadIdx.x;
  if (i >= N_ * 3) return;
  int n = i / 3;
  out[i] = mask[n] ? ac_dn[i] : ac_fs[i];
}

// ---------------------------------------------------------------------------
extern "C" void kernel_launch(void* const* d_in, const int* in_sizes, int n_in,
                              void* d_out, int out_size, void* d_ws,
                              size_t ws_size, hipStream_t stream) {
  const float* x = (const float*)d_in[0];
  const int* zn = (const int*)d_in[1];
  const float* dist = (const float*)d_in[2];
  const int* eidx = (const int*)d_in[3];
  const unsigned char* mask = (const unsigned char*)d_in[4];  // jnp bool (1B)
  const float* wig = (const float*)d_in[5];
  BlockW fs = {(const float*)d_in[6],  (const float*)d_in[7],
               (const float*)d_in[8],  (const float*)d_in[9],
               (const float*)d_in[10], (const float*)d_in[11],
               (const float*)d_in[12], (const float*)d_in[13],
               (const float*)d_in[14], (const float*)d_in[15]};
  BlockW dn = {(const float*)d_in[16], (const float*)d_in[17],
               (const float*)d_in[18], (const float*)d_in[19],
               (const float*)d_in[20], (const float*)d_in[21],
               (const float*)d_in[22], (const float*)d_in[23],
               (const float*)d_in[24], (const float*)d_in[25]};
  const int E = in_sizes[2];  // edge_distance count
  const int N = in_sizes[1];  // atomic_numbers count

  // workspace layout (floats/uints, 4B each): ~4.5 MB total
  float* p = (float*)d_ws;
  float* lg_fs = p; p += (size_t)E * 8;
  float* lg_dn = p; p += (size_t)E * 8;
  unsigned* nm_fs = (unsigned*)p; p += (size_t)N * 8;
  unsigned* nm_dn = (unsigned*)p; p += (size_t)N * 8;
  float* ns_fs = p; p += (size_t)N * 8;
  float* ns_dn = p; p += (size_t)N * 8;
  float* ac_fs = p; p += (size_t)N * 3;
  float* ac_dn = p; p += (size_t)N * 3;

  init_ws<<<(N * 8 + 255) / 256, 256, 0, stream>>>(nm_fs, nm_dn, ns_fs, ns_dn,
                                                   ac_fs, ac_dn, N * 8, N * 3);
  pass_logits<<<(E + 7) / 8, 256, 0, stream>>>(x, zn, dist, eidx, wig, fs, dn,
                                               lg_fs, lg_dn, nm_fs, nm_dn, E);
  pass_soft<<<(E * 8 + 255) / 256, 256, 0, stream>>>(eidx, lg_fs, lg_dn, nm_fs,
                                                     nm_dn, ns_fs, ns_dn, E);
  pass_main<<<E, 128, 0, stream>>>(x, zn, dist, eidx, wig, fs, dn, lg_fs, lg_dn,
                                   ns_fs, ns_dn, ac_fs, ac_dn, E);
  pass_out<<<(N * 3 + 255) / 256, 256, 0, stream>>>(mask, ac_fs, ac_dn,
                                                    (float*)d_out, N);
}